// DetectHead_83253646065693
// MI455X (gfx1250) — compile-verified
//
#include <hip/hip_runtime.h>
#include <hip/hip_bf16.h>

#define N_ANCH 49104
#define NB 16
#define MDET 1000
#define SCORE_THR 0.05f
#define IOU_THR 0.3f
#define CAND_CAP 2048

typedef unsigned int u32x4 __attribute__((ext_vector_type(4)));
typedef int i32x4 __attribute__((ext_vector_type(4)));
typedef int i32x8 __attribute__((ext_vector_type(8)));

struct Ins {
  const float* cls[5];
  const float* reg[5];
  const float* tc[5];
  const float* tr[5];
};

__device__ __forceinline__ void anchor_info(int a, int& l, int& h, int& s, int& loc) {
  if (a < 36864)      { l = 0; h = 192; s = 8;   loc = a; }
  else if (a < 46080) { l = 1; h = 96;  s = 16;  loc = a - 36864; }
  else if (a < 48384) { l = 2; h = 48;  s = 32;  loc = a - 46080; }
  else if (a < 48960) { l = 3; h = 24;  s = 64;  loc = a - 48384; }
  else                { l = 4; h = 12;  s = 128; loc = a - 48960; }
}

// ---------------- Kernel 1: per-anchor max-class score (sigmoid only on the max) ---
// 2D grid: blockIdx.y = batch (no integer division in the hot path).
__global__ void k_scores(Ins in, float* __restrict__ scores) {
  int a = blockIdx.x * blockDim.x + threadIdx.x;
  if (a >= N_ANCH) return;
  int b = blockIdx.y;
  int l, h, s, loc; anchor_info(a, l, h, s, loc);
  int hh = h * h;
  const float* pc = in.cls[l] + (size_t)b * 15 * hh + loc;
  float m = pc[0];
#pragma unroll
  for (int c = 1; c < 15; ++c) {
    float v = pc[(size_t)c * hh];
    m = fmaxf(m, v);
  }
  scores[(size_t)b * N_ANCH + a] = 1.0f / (1.0f + expf(-m));
}

// ---------------- Kernel 2: exact per-batch top-1000 (radix-select + bitonic) ------
__global__ __launch_bounds__(1024) void k_topk(const float* __restrict__ scores,
                                               float* __restrict__ topk_s,
                                               int* __restrict__ topk_i) {
  int b = blockIdx.x;
  int tid = threadIdx.x;
  const float* sc = scores + (size_t)b * N_ANCH;

  __shared__ unsigned hist[256];
  __shared__ unsigned s_pref;
  __shared__ int s_krem;
  __shared__ unsigned s_cnt;
  __shared__ unsigned long long cand[CAND_CAP];

  unsigned pref = 0;
  int krem = MDET;
  // scores are sigmoid outputs in (0,1): positive floats -> uint bit order == float order
  for (int pass = 0; pass < 4; ++pass) {
    int shift = 24 - 8 * pass;
    for (int i = tid; i < 256; i += blockDim.x) hist[i] = 0u;
    __syncthreads();
    for (int i = tid; i < N_ANCH; i += blockDim.x) {
      unsigned key = __float_as_uint(sc[i]);
      bool match = (pass == 0) || ((key >> (shift + 8)) == pref);
      if (match) atomicAdd(&hist[(key >> shift) & 0xFFu], 1u);
    }
    __syncthreads();
    if (tid == 0) {
      int cum = 0, bsel = 0;
      for (int bin = 255; bin >= 0; --bin) {
        int c = (int)hist[bin];
        if (cum + c >= krem) { bsel = bin; break; }
        cum += c;
      }
      s_pref = (pref << 8) | (unsigned)bsel;
      s_krem = krem - cum;
    }
    __syncthreads();
    pref = s_pref;
    krem = s_krem;
    __syncthreads();
  }
  unsigned T = pref;  // exact bit pattern of the 1000th-largest value

  if (tid == 0) s_cnt = 0u;
  __syncthreads();
  for (int i = tid; i < N_ANCH; i += blockDim.x) {
    unsigned key = __float_as_uint(sc[i]);
    if (key >= T) {
      unsigned pos = atomicAdd(&s_cnt, 1u);
      if (pos < CAND_CAP)
        cand[pos] = ((unsigned long long)key << 32) | (unsigned)(~(unsigned)i);
    }
  }
  __syncthreads();
  unsigned cnt = s_cnt; if (cnt > CAND_CAP) cnt = CAND_CAP;
  for (int i = tid; i < CAND_CAP; i += blockDim.x)
    if (i >= (int)cnt) cand[i] = 0ull;  // pads sort last (keys > 0 always)
  __syncthreads();

  // bitonic sort descending on (value, ~index) -> value desc, index asc on ties
  for (int k = 2; k <= CAND_CAP; k <<= 1) {
    for (int jj = k >> 1; jj > 0; jj >>= 1) {
      for (int i = tid; i < CAND_CAP; i += blockDim.x) {
        int ixj = i ^ jj;
        if (ixj > i) {
          unsigned long long va = cand[i], vb = cand[ixj];
          bool desc = ((i & k) == 0);
          if (desc ? (va < vb) : (va > vb)) { cand[i] = vb; cand[ixj] = va; }
        }
      }
      __syncthreads();
    }
  }
  if (tid < MDET) {
    unsigned long long v = cand[tid];
    topk_s[b * MDET + tid] = __uint_as_float((unsigned)(v >> 32));
    topk_i[b * MDET + tid] = (int)(~(unsigned)(v & 0xFFFFFFFFull));
  }
}

// ---------------- Kernel 3: decode class/theta/box for the 16k selected anchors ----
__global__ void k_decode(Ins in, const int* __restrict__ topk_i,
                         float4* __restrict__ boxes4, float* __restrict__ theta_o,
                         float* __restrict__ clsf) {
  int m0 = blockIdx.x * blockDim.x + threadIdx.x;
  if (m0 >= MDET) return;
  int b = blockIdx.y;
  int t = b * MDET + m0;
  int a = topk_i[t];
  int l, h, s, loc; anchor_info(a, l, h, s, loc);
  int hh = h * h;

  const float* pr = in.reg[l] + (size_t)b * 5 * hh + loc;
  __builtin_prefetch(pr, 0, 0);  // global_prefetch_b8

  // class = argmax over 15 logits (first-occurrence ties, matches jnp.argmax)
  const float* pc = in.cls[l] + (size_t)b * 15 * hh + loc;
  float m = pc[0]; int am = 0;
#pragma unroll
  for (int c = 1; c < 15; ++c) {
    float v = pc[(size_t)c * hh];
    if (v > m) { m = v; am = c; }
  }
  clsf[t] = (float)(am + 1);

  // theta = (argmax_tc + 1)*10 + tr
  const float* pt = in.tc[l] + (size_t)b * 18 * hh + loc;
  float mt = pt[0]; int at = 0;
#pragma unroll
  for (int c = 1; c < 18; ++c) {
    float v = pt[(size_t)c * hh];
    if (v > mt) { mt = v; at = c; }
  }
  float tr = in.tr[l][(size_t)b * hh + loc];
  theta_o[t] = (float)(at + 1) * 10.0f + tr;

  int y = loc / h, x = loc % h;
  float fs = (float)s;
  float cx = (float)x * fs + (float)(s >> 1);
  float cy = (float)y * fs + (float)(s >> 1);
  float4 bx;
  bx.x = cx - pr[0] * fs;
  bx.y = cy - pr[(size_t)hh] * fs;
  bx.z = cx + pr[(size_t)2 * hh] * fs;
  bx.w = cy + pr[(size_t)3 * hh] * fs;
  boxes4[t] = bx;
}

// ---------------- Kernel 4: greedy NMS, boxes staged via TDM, scores via async DMA -
__global__ __launch_bounds__(1024) void k_nms(const float4* __restrict__ boxes4,
                                              const float* __restrict__ topk_s,
                                              const float* __restrict__ theta,
                                              const float* __restrict__ clsf,
                                              float* __restrict__ out) {
  int b = blockIdx.x;
  int j = threadIdx.x;

  __shared__ __align__(16) float s_bx[MDET * 4];  // 16 KB box tile
  __shared__ __align__(16) float s_sc[MDET];      // 4 KB score tile
  __shared__ float s_area[MDET];
  __shared__ int s_keep[MDET];

  // --- Tensor Data Mover: one descriptor moves the whole 16 KB box tile to LDS.
  // D# per ISA ch.8: count=1, type=2, data_size=4B, 1D tile of 4000 dwords.
  if (threadIdx.x < 32) {  // wave 0 issues the DMA (EXEC ignored by TDM)
    unsigned long long ga =
        (unsigned long long)(size_t)(boxes4 + (size_t)b * MDET);
    unsigned ldsa = (unsigned)(size_t)(&s_bx[0]);
    u32x4 g0;
    g0.x = 1u;                                            // count=1, user mode
    g0.y = ldsa;                                          // lds_addr
    g0.z = (unsigned)(ga & 0xFFFFFFFFull);                // global_addr[31:0]
    g0.w = (unsigned)((ga >> 32) & 0x1FFFFFFull) | (2u << 30);  // [56:32] + type=2
    i32x8 g1;
    g1[0] = 0x00020000;            // data_size=2 (4 bytes), no pad/iterate/multicast
    g1[1] = (int)(4000u << 16);    // tensor_dim0 = 4000 dwords
    g1[2] = (int)(1u << 16);       // tensor_dim1 = 1
    g1[3] = (int)(4000u << 16);    // tile_dim0 = 4000 dwords
    g1[4] = 0;                     // tile_dim1 = 0 (unused), tile_dim2 = 0
    g1[5] = 4000;                  // tensor_dim0_stride = 4000
    g1[6] = 0;
    g1[7] = 0;
    i32x4 z4 = {0, 0, 0, 0};
#if defined(__clang_major__) && (__clang_major__ >= 23)
    i32x8 z8 = {0, 0, 0, 0, 0, 0, 0, 0};
    __builtin_amdgcn_tensor_load_to_lds(g0, g1, z4, z4, z8, 0);
#else
    __builtin_amdgcn_tensor_load_to_lds(g0, g1, z4, z4, 0);
#endif
    __builtin_amdgcn_s_wait_tensorcnt(0);
  }

  // --- Async global->LDS (ASYNCcnt path) for the per-lane score tile.
  {
    int jj = (j < MDET) ? j : (MDET - 1);
    unsigned long long gsc =
        (unsigned long long)(size_t)(topk_s + (size_t)b * MDET);
    unsigned voff = (unsigned)(jj * 4);
    unsigned ldss = (unsigned)(size_t)(&s_sc[0]) + (unsigned)(jj * 4);
    asm volatile("global_load_async_to_lds_b32 %0, %1, %2"
                 :: "v"(ldss), "v"(voff), "s"(gsc) : "memory");
    asm volatile("s_wait_asynccnt 0" ::: "memory");
  }
  __syncthreads();

  float myscore = 0.f, mytheta = 0.f, mycls = 0.f;
  float4 mybox = {0.f, 0.f, 0.f, 0.f};
  if (j < MDET) {
    int g = b * MDET + j;
    myscore = s_sc[j];
    mytheta = theta[g];
    mycls   = clsf[g];
    mybox.x = s_bx[j * 4 + 0];
    mybox.y = s_bx[j * 4 + 1];
    mybox.z = s_bx[j * 4 + 2];
    mybox.w = s_bx[j * 4 + 3];
    float w  = fmaxf(mybox.z - mybox.x, 0.f);
    float ht = fmaxf(mybox.w - mybox.y, 0.f);
    s_area[j] = w * ht;
    s_keep[j] = (myscore >= SCORE_THR) ? 1 : 0;
  }
  __syncthreads();

  float myarea = (j < MDET) ? s_area[j] : 0.f;
  for (int i = 0; i < MDET; ++i) {
    int ki = s_keep[i];  // only writers of s_keep[i] ran in iterations < i
    if (ki && (j < MDET) && (j > i) && s_keep[j]) {
      float ix1 = s_bx[i * 4 + 0], iy1 = s_bx[i * 4 + 1];
      float ix2 = s_bx[i * 4 + 2], iy2 = s_bx[i * 4 + 3];
      float iw = fmaxf(fminf(ix2, mybox.z) - fmaxf(ix1, mybox.x), 0.f);
      float ih = fmaxf(fminf(iy2, mybox.w) - fmaxf(iy1, mybox.y), 0.f);
      float inter = iw * ih;
      float uni = s_area[i] + myarea - inter;
      float iou = inter / fmaxf(uni, 1e-8f);
      if (iou > IOU_THR) s_keep[j] = 0;
    }
    __syncthreads();
  }

  if (j < MDET) {
    int g = b * MDET + j;
    float k = s_keep[j] ? 1.0f : 0.0f;
    out[g] = myscore * k;                       // scores_post
    out[NB * MDET + g] = mycls * k;             // classes_post (int -> f32 buffer)
    float* ob = out + 2 * NB * MDET + (size_t)g * 5;
    ob[0] = mybox.x * k;
    ob[1] = mybox.y * k;
    ob[2] = mybox.z * k;
    ob[3] = mybox.w * k;
    ob[4] = mytheta * k;                        // boxes_post
  }
}

extern "C" void kernel_launch(void* const* d_in, const int* in_sizes, int n_in,
                              void* d_out, int out_size, void* d_ws, size_t ws_size,
                              hipStream_t stream) {
  (void)in_sizes; (void)n_in; (void)out_size; (void)ws_size;
  Ins in;
  for (int l = 0; l < 5; ++l) {
    in.cls[l] = (const float*)d_in[l];
    in.reg[l] = (const float*)d_in[5 + l];
    in.tc[l]  = (const float*)d_in[10 + l];
    in.tr[l]  = (const float*)d_in[15 + l];
  }
  char* ws = (char*)d_ws;
  // workspace layout (16B aligned sections), total ~3.65 MB
  float*  scores = (float*)(ws + 0);                     // NB*N_ANCH f32
  float4* boxes4 = (float4*)(ws + 3142656);              // NB*MDET float4
  float*  topk_s = (float*)(ws + 3142656 + 256000);      // NB*MDET f32
  float*  theta  = (float*)(ws + 3142656 + 256000 + 64000);
  float*  clsf   = (float*)(ws + 3142656 + 256000 + 128000);
  int*    topk_i = (int*)  (ws + 3142656 + 256000 + 192000);
  float*  out    = (float*)d_out;

  {
    dim3 grid((N_ANCH + 255) / 256, NB);
    k_scores<<<grid, 256, 0, stream>>>(in, scores);
  }
  k_topk<<<NB, 1024, 0, stream>>>(scores, topk_s, topk_i);
  {
    dim3 grid((MDET + 255) / 256, NB);
    k_decode<<<grid, 256, 0, stream>>>(in, topk_i, boxes4, theta, clsf);
  }
  k_nms<<<NB, 1024, 0, stream>>>(boxes4, topk_s, theta, clsf, out);
}